// Time_filter_47330539602310
// MI455X (gfx1250) — compile-verified
//
#include <hip/hip_runtime.h>
#include <cmath>

typedef float v2f __attribute__((ext_vector_type(2)));
typedef float v8f __attribute__((ext_vector_type(8)));

#define HDIM  320
#define WDIM  320
#define HW    102400      // 320*320
#define TDIM  16
#define PLANE 1638400     // 16*320*320 elements per (b,c)
#define NBC   16          // B*C
#define NB    8

// order-preserving float <-> uint encoding for atomic min/max
__device__ __forceinline__ unsigned encf(float f) {
  unsigned u = __float_as_uint(f);
  return (u & 0x80000000u) ? ~u : (u | 0x80000000u);
}
__device__ __forceinline__ float decf(unsigned u) {
  return (u & 0x80000000u) ? __uint_as_float(u & 0x7FFFFFFFu) : __uint_as_float(~u);
}

// ---------------- K0: init accumulators ----------------
__global__ void k0_init(float* red, unsigned* gmn, unsigned* gmx) {
  int i = threadIdx.x;            // 256 threads
  if (i < 64) red[i] = 0.0f;      // 16 * 4 partial sums
  gmn[i] = 0xFFFFFFFFu;           // 256 per-(b,o,t) mins
  gmx[i] = 0u;                    // 256 per-(b,o,t) maxs
}

// ---------------- K1: per-(b,c) sums: x, x^2, around ----------------
// sum(around) = sum over positives of (rh*rw - 1)  (window coverage counts)
__global__ void __launch_bounds__(256) k1_reduce(const float* __restrict__ x,
                                                 float* __restrict__ red) {
  int bc  = blockIdx.y;
  int tid = threadIdx.x;
  const float* xb = x + (size_t)bc * PLANE;
  float sx = 0.f, sx2 = 0.f, sa = 0.f;
  for (int e = blockIdx.x * 256 + tid; e < PLANE; e += gridDim.x * 256) {
    float v = xb[e];
    sx  += v;
    sx2 += v * v;
    if (v > 0.0f) {
      int rem = e % HW;
      int h = rem / WDIM, w = rem % WDIM;
      float rh = (h == 0 || h == HDIM - 1) ? 2.0f : 3.0f;
      float rw = (w == 0 || w == WDIM - 1) ? 2.0f : 3.0f;
      sa += rh * rw - 1.0f;
    }
  }
  __shared__ float s0[256], s1[256], s2[256];
  s0[tid] = sx; s1[tid] = sx2; s2[tid] = sa;
  __syncthreads();
  for (int off = 128; off > 0; off >>= 1) {
    if (tid < off) { s0[tid] += s0[tid + off]; s1[tid] += s1[tid + off]; s2[tid] += s2[tid + off]; }
    __syncthreads();
  }
  if (tid == 0) {
    atomicAdd(&red[bc * 4 + 0], s0[0]);
    atomicAdd(&red[bc * 4 + 1], s1[0]);
    atomicAdd(&red[bc * 4 + 2], s2[0]);
  }
}

// ---------------- K2: stats + dynamic weights -> banded 32x32 matrix ----------------
__global__ void k2_stats(const float* __restrict__ red,
                         const float* __restrict__ fc1,   // (1,2)
                         const float* __restrict__ fc2,   // (4,1)
                         const float* __restrict__ dynw,  // (4,2,2,3)
                         float* __restrict__ stats,       // [16][4]: mu, rs, thr
                         float* __restrict__ Amat,        // [8][32][32]
                         float* __restrict__ bias) {      // [8][32]
  int tid = threadIdx.x;
  __shared__ float s_mu[NBC], s_rs[NBC];
  const float Ninv = 1.0f / (float)PLANE;
  if (tid < NBC) {
    float sx = red[tid * 4 + 0], sx2 = red[tid * 4 + 1], sa = red[tid * 4 + 2];
    float mu  = sx * Ninv;
    float var = sx2 * Ninv - mu * mu;
    float rs  = rsqrtf(var + 1e-5f);
    float thr = 5.0f * sa * Ninv;
    stats[tid * 4 + 0] = mu;
    stats[tid * 4 + 1] = rs;
    stats[tid * 4 + 2] = thr;
    s_mu[tid] = mu; s_rs[tid] = rs;
  }
  __syncthreads();
  if (tid < NB) {
    int b = tid;
    // a_c = mean(xn) over (T,H,W) == (mean(x)-mu)*rs == 0 analytically
    float a0 = 0.0f, a1 = 0.0f;
    float hfc = fmaxf(0.0f, fc1[0] * a0 + fc1[1] * a1);
    float z[4], zmax = -1e30f;
    for (int k = 0; k < 4; k++) { z[k] = fc2[k] * hfc; zmax = fmaxf(zmax, z[k]); }
    float se = 0.0f;
    for (int k = 0; k < 4; k++) { z[k] = __expf(z[k] - zmax); se += z[k]; }
    float w[2][2][3];
    for (int o = 0; o < 2; o++)
      for (int i = 0; i < 2; i++)
        for (int d = 0; d < 3; d++) {
          float acc = 0.0f;
          for (int k = 0; k < 4; k++)
            acc += (z[k] / se) * dynw[((k * 2 + o) * 2 + i) * 3 + d];
          w[o][i][d] = acc;
        }
    float mu_i[2] = { s_mu[b * 2 + 0], s_mu[b * 2 + 1] };
    float rs_i[2] = { s_rs[b * 2 + 0], s_rs[b * 2 + 1] };
    for (int o = 0; o < 2; o++)
      for (int t = 0; t < TDIM; t++) {
        int M = o * 16 + t;
        float bsum = 0.0f;
        for (int i = 0; i < 2; i++)
          for (int tp = 0; tp < TDIM; tp++) {
            int d = tp - t + 1;                    // temporal tap (pad=1)
            float av = (d >= 0 && d < 3) ? w[o][i][d] * rs_i[i] : 0.0f;
            Amat[(b * 32 + M) * 32 + (i * 16 + tp)] = av;
            bsum += av * mu_i[i];
          }
        bias[b * 32 + M] = bsum;                   // out = A'*x - bias
      }
  }
}

// ---------------- K3: WMMA GEMM  out(32 x HW) = A'(32x32) * x(32 x HW) - bias ----------------
__global__ void __launch_bounds__(256) k3_conv(const float* __restrict__ x,
                                               const float* __restrict__ Amat,
                                               const float* __restrict__ bias,
                                               float* __restrict__ out,
                                               unsigned* __restrict__ gmn,
                                               unsigned* __restrict__ gmx) {
  int b    = blockIdx.y;
  int lane = threadIdx.x & 31;
  int wave = threadIdx.x >> 5;
  int lh   = lane >> 4;                 // lane half (0/1)
  int mrow = lane & 15;                 // M (for A) / N pixel (for B,C)
  int p    = (blockIdx.x * 8 + wave) * 16 + mrow;   // pixel index (6400 waves == HW/16 exactly)

  __shared__ unsigned smn[32], smx[32];
  if (threadIdx.x < 32) { smn[threadIdx.x] = 0xFFFFFFFFu; smx[threadIdx.x] = 0u; }
  __syncthreads();

  // C layout: vgpr r, lanes0-15 -> M=r, lanes16-31 -> M=r+8 ; init with -bias
  v8f acc0, acc1;
#pragma unroll
  for (int r = 0; r < 8; r++) {
    acc0[r] = -bias[b * 32 + r + lh * 8];
    acc1[r] = -bias[b * 32 + 16 + r + lh * 8];
  }

  const float* xb = x + (size_t)b * 32 * HW;       // 32 planes (i,t') of HW pixels
  const float* Ab = Amat + b * 1024;

#pragma unroll
  for (int kb = 0; kb < 8; kb++) {
    int k = kb * 4 + lh * 2;                       // A/B layout: half selects K+0/K+2, elem +0/+1
    v2f a0, a1, bb;
    a0.x = Ab[mrow * 32 + k];
    a0.y = Ab[mrow * 32 + k + 1];
    a1.x = Ab[(mrow + 16) * 32 + k];
    a1.y = Ab[(mrow + 16) * 32 + k + 1];
    bb.x = xb[(size_t)k * HW + p];
    bb.y = xb[(size_t)(k + 1) * HW + p];
    acc0 = __builtin_amdgcn_wmma_f32_16x16x4_f32(false, a0, false, bb, (short)0, acc0, false, false);
    acc1 = __builtin_amdgcn_wmma_f32_16x16x4_f32(false, a1, false, bb, (short)0, acc1, false, false);
  }

  float* ob = out + (size_t)b * 32 * HW;           // rows M = o*16+t
#pragma unroll
  for (int r = 0; r < 8; r++) {
    int M0 = r + lh * 8;
    float v0 = tanhf(acc0[r]);
    ob[(size_t)M0 * HW + p] = v0;
    unsigned e0 = encf(v0);
    atomicMin(&smn[M0], e0); atomicMax(&smx[M0], e0);

    int M1 = 16 + r + lh * 8;
    float v1 = tanhf(acc1[r]);
    ob[(size_t)M1 * HW + p] = v1;
    unsigned e1 = encf(v1);
    atomicMin(&smn[M1], e1); atomicMax(&smx[M1], e1);
  }
  __syncthreads();
  if (threadIdx.x < 32) {
    atomicMin(&gmn[b * 32 + threadIdx.x], smn[threadIdx.x]);
    atomicMax(&gmx[b * 32 + threadIdx.x], smx[threadIdx.x]);
  }
}

// ---------------- K4: mask (LDS-tiled 3x3 stencils) + min-max normalize ----------------
__global__ void __launch_bounds__(256) k4_mask(const float* __restrict__ x,
                                               const float* __restrict__ stats,
                                               const unsigned* __restrict__ gmn,
                                               const unsigned* __restrict__ gmx,
                                               float* __restrict__ out) {
  int z  = blockIdx.z;        // = b*32 + c*16 + t
  int t  = z & 15;
  int bc = z >> 4;            // b*2 + c
  int tn = (t == TDIM - 1) ? t : t + 1;   // temporal replication pad
  float thr = stats[bc * 4 + 2];

  int h0 = blockIdx.y * 32, w0 = blockIdx.x * 32;
  __shared__ float bx0[36][36];
  __shared__ float bx1[36][36];
  __shared__ float bs [36][36];
  __shared__ float mm [36][36];

  const float* xt  = x + ((size_t)bc * TDIM + t ) * HW;
  const float* xt1 = x + ((size_t)bc * TDIM + tn) * HW;
  int tid = threadIdx.y * 16 + threadIdx.x;

  for (int idx = tid; idx < 36 * 36; idx += 256) {
    int j = idx / 36, i = idx % 36;
    int gh = h0 + j - 2, gw = w0 + i - 2;
    bool in = (gh >= 0) && (gh < HDIM) && (gw >= 0) && (gw < WDIM);
    float v0 = in ? xt [gh * WDIM + gw] : 0.0f;
    float v1 = in ? xt1[gh * WDIM + gw] : 0.0f;
    bx0[j][i] = (v0 > 0.0f) ? 1.0f : 0.0f;
    bx1[j][i] = (v1 > 0.0f) ? 1.0f : 0.0f;
  }
  __syncthreads();

  for (int idx = tid; idx < 34 * 34; idx += 256) {
    int j = idx / 34 + 1, i = idx % 34 + 1;
    bs[j][i] = bx0[j-1][i-1] + bx0[j-1][i] + bx0[j-1][i+1]
             + bx0[j  ][i-1] + bx0[j  ][i] + bx0[j  ][i+1]
             + bx0[j+1][i-1] + bx0[j+1][i] + bx0[j+1][i+1];
  }
  __syncthreads();

  for (int idx = tid; idx < 34 * 34; idx += 256) {
    int j = idx / 34 + 1, i = idx % 34 + 1;
    float c   = bx0[j][i];
    float around = bs[j][i] - c;
    bool pos   = c > 0.5f;
    bool smask = (around < thr) && pos;
    // tmask: (box3(bx_t)/9) * bx_{t+1} > 1/3  <=>  bx_{t+1}>0 && box3 > 3
    bool tm    = (bx1[j][i] > 0.5f) && (bs[j][i] > 3.0f);
    mm[j][i] = (tm || (!smask && pos)) ? 1.0f : 0.0f;
  }
  __syncthreads();

  unsigned umn = gmn[z], umx = gmx[z];
  float mn = decf(umn), mx = decf(umx);
  float diff = mx - mn;
  float inv = (diff == 0.0f) ? 0.0f : 1.0f / diff;
  size_t obase = (size_t)z * HW;

  for (int py = threadIdx.y; py < 32; py += 16)
    for (int px = threadIdx.x; px < 32; px += 16) {
      int gh = h0 + py, gw = w0 + px;
      int j = py + 2, i = px + 2;
      float msum = mm[j-1][i-1] + mm[j-1][i] + mm[j-1][i+1]
                 + mm[j  ][i-1] + mm[j  ][i] + mm[j  ][i+1]
                 + mm[j+1][i-1] + mm[j+1][i] + mm[j+1][i+1];
      float maskv = msum * (1.0f / 9.0f);
      size_t oi = obase + (size_t)gh * WDIM + gw;
      float pre = out[oi];
      float val = (diff == 0.0f) ? 0.0f : (pre - mn) * inv;
      out[oi] = val * maskv;
    }
}

// ---------------- launch ----------------
extern "C" void kernel_launch(void* const* d_in, const int* in_sizes, int n_in,
                              void* d_out, int out_size, void* d_ws, size_t ws_size,
                              hipStream_t stream) {
  const float* x    = (const float*)d_in[0];
  const float* fc1  = (const float*)d_in[1];
  const float* fc2  = (const float*)d_in[2];
  const float* dynw = (const float*)d_in[3];
  float* out = (float*)d_out;

  char* ws = (char*)d_ws;
  float*    red   = (float*)(ws + 0);       // 64 floats
  float*    stats = (float*)(ws + 256);     // 64 floats
  unsigned* gmn   = (unsigned*)(ws + 512);  // 256
  unsigned* gmx   = (unsigned*)(ws + 1536); // 256
  float*    bias  = (float*)(ws + 2560);    // 256 floats
  float*    Amat  = (float*)(ws + 3584);    // 8192 floats

  k0_init  <<<1, 256, 0, stream>>>(red, gmn, gmx);
  k1_reduce<<<dim3(64, NBC), 256, 0, stream>>>(x, red);
  k2_stats <<<1, 32, 0, stream>>>(red, fc1, fc2, dynw, stats, Amat, bias);
  k3_conv  <<<dim3(HW / 128, NB), 256, 0, stream>>>(x, Amat, bias, out, gmn, gmx);
  k4_mask  <<<dim3(10, 10, NB * 2 * TDIM), dim3(16, 16), 0, stream>>>(x, stats, gmn, gmx, out);
}